// PeerInteractionFeatureExtractor_90400471646753
// MI455X (gfx1250) — compile-verified
//
#include <hip/hip_runtime.h>

#define N_NODES_C 100000
#define N_EDGES_C 3200000
#define IN_FEATS_C 128
#define OUT_FEATS_C 64

typedef float v2f __attribute__((ext_vector_type(2)));
typedef float v8f __attribute__((ext_vector_type(8)));

// ---------------------------------------------------------------------------
// Kernel 1: zero the workspace accumulators (outdeg, indeg, aggs = 3N floats)
// float4-vectorized (3N divisible by 4).
// ---------------------------------------------------------------------------
__global__ void gc_zero_kernel(float4* __restrict__ p, int n4) {
    int i = blockIdx.x * blockDim.x + threadIdx.x;
    if (i < n4) p[i] = make_float4(0.f, 0.f, 0.f, 0.f);
}

// ---------------------------------------------------------------------------
// Kernel 2: degree counting. 4 edges/thread via int4 loads (coalesced
// 512B/wave requests), float atomics (exact for counts < 2^24).
// ---------------------------------------------------------------------------
__global__ void gc_degree_kernel(const int4* __restrict__ src4,
                                 const int4* __restrict__ dst4,
                                 float* __restrict__ outdeg,
                                 float* __restrict__ indeg, int E4) {
    int i = blockIdx.x * blockDim.x + threadIdx.x;
    if (i < E4) {
        int4 sv = src4[i];
        int4 dv = dst4[i];
        atomicAdd(&outdeg[sv.x], 1.0f);
        atomicAdd(&outdeg[sv.y], 1.0f);
        atomicAdd(&outdeg[sv.z], 1.0f);
        atomicAdd(&outdeg[sv.w], 1.0f);
        atomicAdd(&indeg[dv.x], 1.0f);
        atomicAdd(&indeg[dv.y], 1.0f);
        atomicAdd(&indeg[dv.z], 1.0f);
        atomicAdd(&indeg[dv.w], 1.0f);
    }
}

// ---------------------------------------------------------------------------
// Kernel 3: s[v] = mean_j( (outnorm[v] * X[v,:]) @ W )[j]
// One wave (32 lanes) computes a 16-row tile with fp32 WMMA (16x16x4),
// 4 column tiles of 16, K swept 128/4 = 32 steps. W staged in LDS once/block.
// C/D layout: VGPR r holds M=r (lanes 0-15) and M=r+8 (lanes 16-31), N=lane%16.
// A layout (16x4 f32): lane l, vgpr v -> A[M = l%16, K = 2*(l/16) + v].
// B layout (4x16 f32): lane l, vgpr v -> B[K = 2*v + l/16,  N = l%16].
// ---------------------------------------------------------------------------
__global__ __launch_bounds__(64)
void gc_gemm_mean_kernel(const float* __restrict__ X,
                         const float* __restrict__ W,
                         const float* __restrict__ outdeg,
                         float* __restrict__ s) {
    __shared__ float Wl[IN_FEATS_C * OUT_FEATS_C];  // 32 KB

    const int tid = threadIdx.x;

    // Stage W into LDS, float4-coalesced (8192 floats = 2048 float4s).
    {
        const float4* Wv  = (const float4*)W;
        float4*       Wlv = (float4*)Wl;
        for (int i = tid; i < (IN_FEATS_C * OUT_FEATS_C) / 4; i += 64)
            Wlv[i] = Wv[i];
    }
    __syncthreads();

    const int lane = tid & 31;
    const int wave = tid >> 5;
    const int tile = blockIdx.x * 2 + wave;       // 3125 blocks * 2 = 6250 tiles
    const int mrow = lane & 15;                   // M within tile
    const int half = lane >> 4;                   // K-pair selector
    const int row  = tile * 16 + mrow;            // global node id (exact fit)

    const float dn = rsqrtf(fmaxf(outdeg[row], 1.0f));  // out_norm for this row
    const float* __restrict__ xrow = X + (size_t)row * IN_FEATS_C;

    v8f c0 = {}, c1 = {}, c2 = {}, c3 = {};

    #pragma unroll 4
    for (int k0 = 0; k0 < IN_FEATS_C / 4; ++k0) {
        // A fragment: K = k0*4 + 2*half + {0,1}, pre-scaled by out_norm.
        float2 av = *(const float2*)(xrow + k0 * 4 + 2 * half);
        v2f a;
        a.x = av.x * dn;
        a.y = av.y * dn;

        const int kb = k0 * 4 + half;  // base K for B vgpr0; vgpr1 is +2
        // B fragments for the four 16-column tiles.
        v2f b0, b1, b2, b3;
        b0.x = Wl[(kb + 0) * OUT_FEATS_C + mrow +  0];
        b0.y = Wl[(kb + 2) * OUT_FEATS_C + mrow +  0];
        b1.x = Wl[(kb + 0) * OUT_FEATS_C + mrow + 16];
        b1.y = Wl[(kb + 2) * OUT_FEATS_C + mrow + 16];
        b2.x = Wl[(kb + 0) * OUT_FEATS_C + mrow + 32];
        b2.y = Wl[(kb + 2) * OUT_FEATS_C + mrow + 32];
        b3.x = Wl[(kb + 0) * OUT_FEATS_C + mrow + 48];
        b3.y = Wl[(kb + 2) * OUT_FEATS_C + mrow + 48];

        c0 = __builtin_amdgcn_wmma_f32_16x16x4_f32(false, a, false, b0, (short)0, c0, false, false);
        c1 = __builtin_amdgcn_wmma_f32_16x16x4_f32(false, a, false, b1, (short)0, c1, false, false);
        c2 = __builtin_amdgcn_wmma_f32_16x16x4_f32(false, a, false, b2, (short)0, c2, false, false);
        c3 = __builtin_amdgcn_wmma_f32_16x16x4_f32(false, a, false, b3, (short)0, c3, false, false);
    }

    // Row-wise mean over all 64 output columns: sum the 4 column tiles,
    // then reduce across the 16 lanes of each half-wave (xor masks < 16
    // stay within a half).
    const float inv = 1.0f / (float)OUT_FEATS_C;
    #pragma unroll
    for (int r = 0; r < 8; ++r) {
        float v = c0[r] + c1[r] + c2[r] + c3[r];
        v += __shfl_xor(v, 1, 32);
        v += __shfl_xor(v, 2, 32);
        v += __shfl_xor(v, 4, 32);
        v += __shfl_xor(v, 8, 32);
        if ((lane & 15) == 0)
            s[tile * 16 + r + half * 8] = v * inv;  // lane0 -> M=r, lane16 -> M=r+8
    }
}

// ---------------------------------------------------------------------------
// Kernel 4: per-edge scalar scatter: aggs[dst] += s[src]  (one atomic/edge)
// 4 edges/thread via int4 loads; s and aggs are 400 KB each -> L2 resident.
// ---------------------------------------------------------------------------
__global__ void gc_scatter_kernel(const int4* __restrict__ src4,
                                  const int4* __restrict__ dst4,
                                  const float* __restrict__ s,
                                  float* __restrict__ aggs, int E4) {
    int i = blockIdx.x * blockDim.x + threadIdx.x;
    if (i < E4) {
        int4 sv = src4[i];
        int4 dv = dst4[i];
        float s0 = s[sv.x];
        float s1 = s[sv.y];
        float s2 = s[sv.z];
        float s3 = s[sv.w];
        atomicAdd(&aggs[dv.x], s0);
        atomicAdd(&aggs[dv.y], s1);
        atomicAdd(&aggs[dv.z], s2);
        atomicAdd(&aggs[dv.w], s3);
    }
}

// ---------------------------------------------------------------------------
// Kernel 5: out[i] = aggs[i] * rsqrt(max(indeg,1)) + mean(b)
// ---------------------------------------------------------------------------
__global__ void gc_finalize_kernel(const float* __restrict__ aggs,
                                   const float* __restrict__ indeg,
                                   const float* __restrict__ b,
                                   float* __restrict__ out, int N) {
    __shared__ float bmean;
    if (threadIdx.x == 0) {
        float t = 0.0f;
        #pragma unroll
        for (int j = 0; j < OUT_FEATS_C; ++j) t += b[j];
        bmean = t * (1.0f / (float)OUT_FEATS_C);
    }
    __syncthreads();
    int i = blockIdx.x * blockDim.x + threadIdx.x;
    if (i < N)
        out[i] = aggs[i] * rsqrtf(fmaxf(indeg[i], 1.0f)) + bmean;
}

// ---------------------------------------------------------------------------
extern "C" void kernel_launch(void* const* d_in, const int* in_sizes, int n_in,
                              void* d_out, int out_size, void* d_ws, size_t ws_size,
                              hipStream_t stream) {
    const float* X   = (const float*)d_in[0];  // [100000,128]
    const float* W   = (const float*)d_in[1];  // [128,64]
    const float* b   = (const float*)d_in[2];  // [64]
    const int4*  src = (const int4*)d_in[3];   // [3.2M] viewed as int4
    const int4*  dst = (const int4*)d_in[4];   // [3.2M] viewed as int4
    float*       out = (float*)d_out;          // [100000]

    float* ws     = (float*)d_ws;
    float* outdeg = ws;                        // [N]
    float* indeg  = ws + N_NODES_C;            // [N]
    float* aggs   = ws + 2 * N_NODES_C;        // [N]
    float* s      = ws + 3 * N_NODES_C;        // [N]

    const int N  = N_NODES_C;
    const int E4 = N_EDGES_C / 4;              // 800000, exact

    // 1) zero accumulators (outdeg, indeg, aggs) every call — graph-replay safe
    gc_zero_kernel<<<((3 * N / 4) + 255) / 256, 256, 0, stream>>>((float4*)ws, 3 * N / 4);

    // 2) degrees (4 edges/thread)
    gc_degree_kernel<<<(E4 + 255) / 256, 256, 0, stream>>>(src, dst, outdeg, indeg, E4);

    // 3) fused normalized-GEMM + feature-mean -> s[v]   (WMMA fp32)
    gc_gemm_mean_kernel<<<N / 32, 64, 0, stream>>>(X, W, outdeg, s);

    // 4) one-atomic-per-edge aggregation (4 edges/thread)
    gc_scatter_kernel<<<(E4 + 255) / 256, 256, 0, stream>>>(src, dst, s, aggs, E4);

    // 5) finalize
    gc_finalize_kernel<<<(N + 255) / 256, 256, 0, stream>>>(aggs, indeg, b, out, N);
}